// weighted_Loss_79809082295091
// MI455X (gfx1250) — compile-verified
//
#include <hip/hip_runtime.h>

// Weighted MSE loss, [B=1024, T=8192, C=2] fp32 -> (loss_a, loss_k).
// Memory-bound streaming reduction: 128 MiB read @ 23.3 TB/s => ~5.8us floor.
// Cross-lane wave reduction implemented with V_WMMA_F32_16X16X4_F32 (B = ones):
// D[m][n] = sum_k A[m][k]; summing all D rows gives the full 32-lane sum
// regardless of the exact (lane,VGPR)->(m,k) mapping.

typedef __attribute__((ext_vector_type(2))) float v2f;
typedef __attribute__((ext_vector_type(8))) float v8f;

__device__ __forceinline__ float wave_reduce_wmma(float v) {
    v2f a; a.x = v;    a.y = 0.0f;   // lane partial in A, second slot zero
    v2f b; b.x = 1.0f; b.y = 1.0f;   // all-ones B => D[m][n] = sum_k A[m][k]
    v8f c = {};
    // 8 args: (neg_a, A, neg_b, B, c_mod, C, reuse_a, reuse_b)
    c = __builtin_amdgcn_wmma_f32_16x16x4_f32(false, a, false, b, (short)0, c,
                                              false, false);
    // Per lane: sum its 8 D rows (half of M), then fold the other half-wave.
    float r = ((c[0] + c[1]) + (c[2] + c[3])) + ((c[4] + c[5]) + (c[6] + c[7]));
    r += __shfl_xor(r, 16, 32);
    return r;  // all lanes now hold the wave-wide sum
}

__global__ __launch_bounds__(256) void wloss_partial_kernel(
    const float* __restrict__ pred, const float* __restrict__ actual,
    float2* __restrict__ partials, unsigned nPairs, unsigned pairsPerRow)
{
    const unsigned stride = gridDim.x * blockDim.x;
    const unsigned gtid   = blockIdx.x * blockDim.x + threadIdx.x;
    const unsigned lane   = threadIdx.x & 31u;

    float sum_a = 0.0f, sum_k = 0.0f;

    // Pair p = timesteps (2p, 2p+1): pred4/ac4 = [a0, k0, a1, k1].
    // pairsPerRow is a multiple of 32 -> a wave never straddles a row, and
    // row-start pairs always land on lane 0.
    #pragma unroll 4
    for (unsigned p = gtid; p < nPairs; p += stride) {
        const float4 pr = ((const float4*)pred)[p];
        const float4 ac = ((const float4*)actual)[p];

        // actual at timestep 2p-1 lives in lane L-1's (z,w); fetch via shuffle.
        float pa = __shfl_up(ac.z, 1, 32);
        float pk = __shfl_up(ac.w, 1, 32);
        const bool rowStart = (p % pairsPerRow) == 0u;
        if (lane == 0u && !rowStart) {
            // wave-boundary: reload 8 bytes (tail of previous pair)
            const float2 pv = *(const float2*)(actual + 4ull * (unsigned long long)p - 2ull);
            pa = pv.x; pk = pv.y;
        }

        const float w0 = (!rowStart && ((pa != ac.x) || (pk != ac.y))) ? 2.0f : 1.0f;
        const float w1 = ((ac.x != ac.z) || (ac.y != ac.w)) ? 2.0f : 1.0f;

        const float ea0 = pr.x - ac.x, ek0 = pr.y - ac.y;
        const float ea1 = pr.z - ac.z, ek1 = pr.w - ac.w;
        sum_a = fmaf(w0, ea0 * ea0, fmaf(w1, ea1 * ea1, sum_a));
        sum_k = fmaf(w0, ek0 * ek0, fmaf(w1, ek1 * ek1, sum_k));
    }

    // Wave reduction via WMMA (EXEC is all-ones here: uniform control flow).
    const float wa = wave_reduce_wmma(sum_a);
    const float wk = wave_reduce_wmma(sum_k);

    __shared__ float2 smem[8];
    const unsigned warp = threadIdx.x >> 5;
    if (lane == 0u) smem[warp] = make_float2(wa, wk);
    __syncthreads();
    if (threadIdx.x == 0u) {
        float2 t = smem[0];
        #pragma unroll
        for (int i = 1; i < 8; ++i) { t.x += smem[i].x; t.y += smem[i].y; }
        partials[blockIdx.x] = t;  // deterministic per-block partial
    }
}

__global__ __launch_bounds__(256) void wloss_final_kernel(
    const float2* __restrict__ partials, unsigned nBlocks,
    float* __restrict__ out, float invN)
{
    float sa = 0.0f, sk = 0.0f;
    for (unsigned i = threadIdx.x; i < nBlocks; i += blockDim.x) {
        const float2 t = partials[i];
        sa += t.x; sk += t.y;
    }
    sa = wave_reduce_wmma(sa);
    sk = wave_reduce_wmma(sk);

    __shared__ float2 smem[8];
    const unsigned lane = threadIdx.x & 31u, warp = threadIdx.x >> 5;
    if (lane == 0u) smem[warp] = make_float2(sa, sk);
    __syncthreads();
    if (threadIdx.x == 0u) {
        float2 t = smem[0];
        #pragma unroll
        for (int i = 1; i < 8; ++i) { t.x += smem[i].x; t.y += smem[i].y; }
        out[0] = t.x * invN;   // loss_a
        out[1] = t.y * invN;   // loss_k
    }
}

extern "C" void kernel_launch(void* const* d_in, const int* in_sizes, int n_in,
                              void* d_out, int out_size, void* d_ws, size_t ws_size,
                              hipStream_t stream) {
    const float* pred   = (const float*)d_in[0];
    const float* actual = (const float*)d_in[1];

    const unsigned total  = (unsigned)in_sizes[0]; // B*T*C = 16,777,216
    const unsigned nPairs = total / 4u;            // 4,194,304 timestep-pairs
    const unsigned T = 8192u;                      // reference shape
    const unsigned pairsPerRow = T / 2u;           // 4096 (multiple of 32)

    const int threads = 256;
    const int blocks  = 1024;                      // 8192 waves; 16 pairs/thread

    float2* partials = (float2*)d_ws;              // 1024 * 8B = 8 KiB scratch

    wloss_partial_kernel<<<blocks, threads, 0, stream>>>(
        pred, actual, partials, nPairs, pairsPerRow);

    const float invN = 1.0f / (float)(total / 2u); // mean over B*T
    wloss_final_kernel<<<1, 256, 0, stream>>>(
        partials, (unsigned)blocks, (float*)d_out, invN);
}